// GCN_41867341201638
// MI455X (gfx1250) — compile-verified
//
#include <hip/hip_runtime.h>
#include <math.h>

#define HDIM 16
#define NCLS 4
#define NGRAPH 256

typedef __attribute__((ext_vector_type(2))) float v2f;
typedef __attribute__((ext_vector_type(8))) float v8f;

// ---------------- degree / normalization ----------------

__global__ void deg_init_kernel(float* __restrict__ deg, int n) {
  int i = blockIdx.x * blockDim.x + threadIdx.x;
  if (i < n) deg[i] = 1.0f;  // self-loop contribution
}

__global__ void deg_edge_kernel(const long long* __restrict__ dst,
                                float* __restrict__ deg, int e) {
  int i = blockIdx.x * blockDim.x + threadIdx.x;
  if (i < e) atomicAdd(&deg[(int)dst[i]], 1.0f);
}

__global__ void dinv_kernel(float* __restrict__ deg, int n) {
  int i = blockIdx.x * blockDim.x + threadIdx.x;
  if (i < n) deg[i] = rsqrtf(deg[i]);  // deg >= 1 always (self-loops)
}

// ---------------- dense transform via V_WMMA_F32_16X16X4_F32 ----------------
// One wave handles a 16-node tile: t = hin(16x16) @ W(16x16), K split 4x4.
// u[node,:] = t[node,:] * dinv[node]   (optionally relu(hin) on load)

__global__ void transform_wmma_kernel(const float* __restrict__ hin,
                                      const float* __restrict__ W,   // [16][16] (in,out)
                                      const float* __restrict__ dinv,
                                      float* __restrict__ u,
                                      int n_tiles, int relu_in) {
  int wave = (int)((blockIdx.x * (unsigned)blockDim.x + threadIdx.x) >> 5);
  int lane = threadIdx.x & 31;
  if (wave >= n_tiles) return;           // wave-uniform exit: EXEC stays all-1s

  const int m  = lane & 15;              // A row / B col / D col
  const int kh = lane >> 4;              // half-wave selects K pair
  const int node = wave * 16 + m;
  const float* __restrict__ row = hin + (long)node * HDIM;

  v8f c = {};
#pragma unroll
  for (int k0 = 0; k0 < 16; k0 += 4) {
    const int k = k0 + 2 * kh;
    float a0 = row[k];
    float a1 = row[k + 1];
    if (relu_in) { a0 = fmaxf(a0, 0.0f); a1 = fmaxf(a1, 0.0f); }
    v2f a, b;
    a.x = a0;                 a.y = a1;                   // A[m][k], A[m][k+1]
    b.x = W[k * HDIM + m];    b.y = W[(k + 1) * HDIM + m]; // B[k][n], B[k+1][n]
    c = __builtin_amdgcn_wmma_f32_16x16x4_f32(
        /*neg_a=*/false, a, /*neg_b=*/false, b,
        /*c_mod=*/(short)0, c, /*reuse_a=*/false, /*reuse_b=*/false);
  }

  const int base = wave * 16;
#pragma unroll
  for (int j = 0; j < 8; ++j) {
    const int r = base + j + 8 * kh;                     // D row
    u[(long)r * HDIM + m] = c[j] * dinv[r];              // col n == m
  }
}

// acc[i,f] = u[i,f]*dinv[i] + b[f]   (self-loop term + bias)
__global__ void acc_init_kernel(const float* __restrict__ u,
                                const float* __restrict__ dinv,
                                const float* __restrict__ bias,
                                float* __restrict__ acc, int n) {
  int idx = blockIdx.x * blockDim.x + threadIdx.x;
  if (idx < n * HDIM) {
    int node = idx >> 4, f = idx & 15;
    acc[idx] = u[idx] * dinv[node] + bias[f];
  }
}

// one thread per (edge, feature): acc[dst,f] += u[src,f]*dinv[dst]
__global__ void edge_scatter_kernel(const long long* __restrict__ ei,  // [2][E] flat
                                    const float* __restrict__ u,
                                    const float* __restrict__ dinv,
                                    float* __restrict__ acc, int e) {
  long long idx = (long long)blockIdx.x * blockDim.x + threadIdx.x;
  if (idx >= (long long)e * HDIM) return;
  int eid = (int)(idx >> 4);
  int f   = (int)(idx & 15);
  int s = (int)ei[eid];
  int d = (int)ei[(long long)e + eid];
  atomicAdd(&acc[(long)d * HDIM + f], u[(long)s * HDIM + f] * dinv[d]);
}

// ---------------- pooling + head ----------------

__global__ void pool_zero_kernel(float* __restrict__ sums, float* __restrict__ cnts) {
  int i = blockIdx.x * blockDim.x + threadIdx.x;
  if (i < NGRAPH * HDIM) sums[i] = 0.0f;
  if (i < NGRAPH) cnts[i] = 0.0f;
}

__global__ void pool_kernel(const float* __restrict__ h,
                            const long long* __restrict__ batch,
                            float* __restrict__ sums, float* __restrict__ cnts, int n) {
  int idx = blockIdx.x * blockDim.x + threadIdx.x;
  if (idx < n * HDIM) {
    int node = idx >> 4, f = idx & 15;
    int g = (int)batch[node];
    atomicAdd(&sums[g * HDIM + f], h[idx]);
    if (f == 0) atomicAdd(&cnts[g], 1.0f);
  }
}

__global__ void head_kernel(const float* __restrict__ sums,
                            const float* __restrict__ cnts,
                            const float* __restrict__ Wlin,   // [16][4]
                            const float* __restrict__ blin,   // [4]
                            float* __restrict__ out) {
  int g = blockIdx.x * blockDim.x + threadIdx.x;
  if (g >= NGRAPH) return;
  float inv = 1.0f / fmaxf(cnts[g], 1.0f);
  float logit[NCLS];
#pragma unroll
  for (int c = 0; c < NCLS; ++c) logit[c] = blin[c];
#pragma unroll
  for (int f = 0; f < HDIM; ++f) {
    float p = sums[g * HDIM + f] * inv;
#pragma unroll
    for (int c = 0; c < NCLS; ++c) logit[c] = fmaf(p, Wlin[f * NCLS + c], logit[c]);
  }
  float mx = logit[0];
#pragma unroll
  for (int c = 1; c < NCLS; ++c) mx = fmaxf(mx, logit[c]);
  float s = 0.0f;
#pragma unroll
  for (int c = 0; c < NCLS; ++c) s += expf(logit[c] - mx);
  float lse = mx + logf(s);
#pragma unroll
  for (int c = 0; c < NCLS; ++c) out[g * NCLS + c] = logit[c] - lse;
}

// ---------------- launch ----------------

extern "C" void kernel_launch(void* const* d_in, const int* in_sizes, int n_in,
                              void* d_out, int out_size, void* d_ws, size_t ws_size,
                              hipStream_t stream) {
  const float*     x    = (const float*)d_in[0];
  const float*     W1   = (const float*)d_in[1];
  const float*     b1   = (const float*)d_in[2];
  const float*     W2   = (const float*)d_in[3];
  const float*     b2   = (const float*)d_in[4];
  const float*     W3   = (const float*)d_in[5];
  const float*     b3   = (const float*)d_in[6];
  const float*     Wlin = (const float*)d_in[7];
  const float*     blin = (const float*)d_in[8];
  const long long* ei   = (const long long*)d_in[9];   // int64 [2][E]
  const long long* bat  = (const long long*)d_in[10];  // int64 [N]

  const int N = in_sizes[0] / HDIM;      // 200000
  const int E = in_sizes[9] / 2;         // 3200000
  const int n_tiles = N / HDIM;          // 12500 (N is a multiple of 16)

  // workspace layout (floats)
  float* ws   = (float*)d_ws;
  float* dinv = ws;                       // N
  float* u    = dinv + N;                 // N*16
  float* bufA = u + (long)N * HDIM;       // N*16
  float* bufB = bufA + (long)N * HDIM;    // N*16
  float* sums = bufB + (long)N * HDIM;    // 256*16
  float* cnts = sums + NGRAPH * HDIM;     // 256

  const int T = 256;
  const int gN   = (N + T - 1) / T;
  const int gE   = (E + T - 1) / T;
  const int gNF  = (N * HDIM + T - 1) / T;
  const long long totEF = (long long)E * HDIM;
  const int gEF  = (int)((totEF + T - 1) / T);
  const int gTil = (n_tiles + (T / 32) - 1) / (T / 32);  // 8 waves / block

  // --- symmetric normalization ---
  deg_init_kernel<<<gN, T, 0, stream>>>(dinv, N);
  deg_edge_kernel<<<gE, T, 0, stream>>>(ei + E, dinv, E);   // dst row
  dinv_kernel<<<gN, T, 0, stream>>>(dinv, N);

  // --- layer 1: x -> bufA ---
  transform_wmma_kernel<<<gTil, T, 0, stream>>>(x, W1, dinv, u, n_tiles, 0);
  acc_init_kernel<<<gNF, T, 0, stream>>>(u, dinv, b1, bufA, N);
  edge_scatter_kernel<<<gEF, T, 0, stream>>>(ei, u, dinv, bufA, E);

  // --- layer 2: relu(bufA) -> bufB ---
  transform_wmma_kernel<<<gTil, T, 0, stream>>>(bufA, W2, dinv, u, n_tiles, 1);
  acc_init_kernel<<<gNF, T, 0, stream>>>(u, dinv, b2, bufB, N);
  edge_scatter_kernel<<<gEF, T, 0, stream>>>(ei, u, dinv, bufB, E);

  // --- layer 3: relu(bufB) -> bufA (no relu on output) ---
  transform_wmma_kernel<<<gTil, T, 0, stream>>>(bufB, W3, dinv, u, n_tiles, 1);
  acc_init_kernel<<<gNF, T, 0, stream>>>(u, dinv, b3, bufA, N);
  edge_scatter_kernel<<<gEF, T, 0, stream>>>(ei, u, dinv, bufA, E);

  // --- mean pool + linear head + log_softmax ---
  pool_zero_kernel<<<(NGRAPH * HDIM + T - 1) / T, T, 0, stream>>>(sums, cnts);
  pool_kernel<<<gNF, T, 0, stream>>>(bufA, bat, sums, cnts, N);
  head_kernel<<<(NGRAPH + T - 1) / T, T, 0, stream>>>(sums, cnts, Wlin, blin,
                                                      (float*)d_out);
}